// AttentionUpscaling_54614804136667
// MI455X (gfx1250) — compile-verified
//
#include <hip/hip_runtime.h>
#include <hip/hip_bf16.h>
#include <math.h>

#define BB   2
#define NN   4096
#define DD   1024
#define EE   128
#define HH   64
#define DIN  256
#define KK   16

typedef float v2f __attribute__((ext_vector_type(2)));
typedef float v4f __attribute__((ext_vector_type(4)));
typedef float v8f __attribute__((ext_vector_type(8)));

// ---------------------------------------------------------------------------
// Kernel 0: interleave weights into k-pair layout  W[k][e] -> Wi[kp][e][2]
// so each WMMA B-operand fetch is one aligned 8B load.
// ---------------------------------------------------------------------------
__global__ __launch_bounds__(256) void interleave_w(
    const float* __restrict__ Wk, const float* __restrict__ Wq,
    const float* __restrict__ W1,
    float* __restrict__ Wki, float* __restrict__ Wqi, float* __restrict__ W1i)
{
    int gid = blockIdx.x * 256 + threadIdx.x;          // 16384 per tensor
    if (blockIdx.y == 0) {
        int e = gid & 127, kp = gid >> 7;              // kp < 128
        Wki[gid * 2 + 0] = Wk[(2 * kp) * EE + e];
        Wki[gid * 2 + 1] = Wk[(2 * kp + 1) * EE + e];
    } else if (blockIdx.y == 1) {
        int e = gid & 127, kp = gid >> 7;
        Wqi[gid * 2 + 0] = Wq[(2 * kp) * EE + e];
        Wqi[gid * 2 + 1] = Wq[(2 * kp + 1) * EE + e];
    } else {
        int e = gid & 63, kp = gid >> 6;               // kp < 256 (W1[0:512])
        W1i[gid * 2 + 0] = W1[(2 * kp) * HH + e];
        W1i[gid * 2 + 1] = W1[(2 * kp + 1) * HH + e];
    }
}

// ---------------------------------------------------------------------------
// Kernel 1: adaptive avg-pool 8x8 -> 4x4, k-pair interleaved output:
// pool[b][kp][n][2]  (element (k,n) at ((b*128+kp)*N+n)*2 + k%2)
// ---------------------------------------------------------------------------
__global__ __launch_bounds__(256) void pool_kernel(
    const float* __restrict__ hr, const float* __restrict__ base,
    float* __restrict__ poolK, float* __restrict__ poolQ)
{
    const float* src = blockIdx.y ? base : hr;
    float*       dst = blockIdx.y ? poolQ : poolK;
    int gid  = blockIdx.x * 256 + threadIdx.x;     // B*128*N threads
    int n    = gid & (NN - 1);
    int rest = gid >> 12;
    int kp   = rest & 127;
    int b    = rest >> 7;
    const float* p = src + (size_t)b * DD * NN + n;
    v2f s;
    #pragma unroll
    for (int j = 0; j < 2; ++j) {
        int dout = 2 * kp + j;
        int c = dout >> 4, py = (dout >> 2) & 3, px = dout & 3;
        float acc = 0.f;
        #pragma unroll
        for (int sy = 0; sy < 2; ++sy)
          #pragma unroll
          for (int sx = 0; sx < 2; ++sx) {
            int din = c * 64 + (2 * py + sy) * 8 + (2 * px + sx);
            acc += p[(size_t)din * NN];
          }
        s[j] = 0.25f * acc;
    }
    *(v2f*)&dst[(((size_t)b * 128 + kp) * NN + n) * 2] = s;
}

// ---------------------------------------------------------------------------
// Kernel 2: encoder GEMM  (8192 x 256) @ (256 x 128) + bias, for q and k.
// One wave -> one 16x16 tile via V_WMMA_F32_16X16X4_F32; all operand loads b64.
// ---------------------------------------------------------------------------
__global__ __launch_bounds__(256) void encode_gemm(
    const float* __restrict__ poolK, const float* __restrict__ poolQ,
    const float* __restrict__ Wki, const float* __restrict__ bk,
    const float* __restrict__ Wqi, const float* __restrict__ bq,
    float* __restrict__ kEmb, float* __restrict__ qEmb)
{
    const float *A, *Wi, *bias; float* out;
    if (blockIdx.y == 0) { A = poolK; Wi = Wki; bias = bk; out = kEmb; }
    else                 { A = poolQ; Wi = Wqi; bias = bq; out = qEmb; }

    int bn0  = blockIdx.x * 16;
    int b    = bn0 >> 12;
    int n0   = bn0 & (NN - 1);
    int lane = threadIdx.x & 31;
    int wave = threadIdx.x >> 5;         // 8 waves -> E=128 cols
    int half = lane >> 4;
    int ml   = lane & 15;
    int e0   = wave * 16;

    const float* Ab = A + ((size_t)b * 128) * NN * 2 + (size_t)(n0 + ml) * 2;
    const float* Wb = Wi + (size_t)(e0 + ml) * 2;
    v8f acc = {};
    for (int kk = 0; kk < DIN / 4; ++kk) {
        int kp = kk * 2 + half;          // k-pair index for this lane-half
        v2f a = *(const v2f*)(Ab + (size_t)kp * NN * 2);
        v2f w = *(const v2f*)(Wb + (size_t)kp * EE * 2);
        acc = __builtin_amdgcn_wmma_f32_16x16x4_f32(false, a, false, w,
                                                    (short)0, acc, false, false);
    }
    #pragma unroll
    for (int r = 0; r < 8; ++r) {        // C: VGPR r holds rows r and r+8
        int row = r + 8 * half;
        int e   = e0 + ml;
        out[(size_t)(bn0 + row) * EE + e] = acc[r] + bias[e];
    }
}

// ---------------------------------------------------------------------------
// Kernel 3: top-16 per 4096-wide attention row. Register candidates +
// wave32 shfl_xor argmax + 8-entry cross-wave merge. Non-temporal row loads.
// ---------------------------------------------------------------------------
__global__ __launch_bounds__(256) void topk_kernel(
    const float* __restrict__ attn, float* __restrict__ prior, int* __restrict__ idx)
{
    __shared__ float rv[8];
    __shared__ int   ri[8];
    __shared__ int   win;
    int bn   = blockIdx.x;
    int tid  = threadIdx.x;
    int lane = tid & 31, wave = tid >> 5;
    const float* row = attn + (size_t)bn * NN;
    float v[16];
    #pragma unroll
    for (int i = 0; i < 16; ++i) v[i] = __builtin_nontemporal_load(&row[tid + 256 * i]);

    for (int it = 0; it < KK; ++it) {
        float best = -INFINITY; int bi = tid;
        #pragma unroll
        for (int i = 0; i < 16; ++i)
            if (v[i] > best) { best = v[i]; bi = tid + 256 * i; }
        #pragma unroll
        for (int off = 16; off > 0; off >>= 1) {       // wave32 argmax
            float ov = __shfl_xor(best, off, 32);
            int   oi = __shfl_xor(bi,   off, 32);
            if (ov > best) { best = ov; bi = oi; }
        }
        if (lane == 0) { rv[wave] = best; ri[wave] = bi; }
        __syncthreads();
        if (tid == 0) {
            float m = rv[0]; int mi = ri[0];
            #pragma unroll
            for (int j = 1; j < 8; ++j)
                if (rv[j] > m) { m = rv[j]; mi = ri[j]; }
            prior[(size_t)bn * KK + it] = m;
            idx[(size_t)bn * KK + it]   = mi;
            win = mi;
        }
        __syncthreads();
        int w = win;
        if ((w & 255) == tid) v[w >> 8] = -INFINITY;   // remove winner
        __syncthreads();
    }
}

// ---------------------------------------------------------------------------
// Kernel 4: transpose hr_hf (B,D,N) -> hfT (B,N,D) so gathers are 4KB rows.
// ---------------------------------------------------------------------------
__global__ __launch_bounds__(256) void transpose_hf(
    const float* __restrict__ hr, float* __restrict__ hfT)
{
    __shared__ float tile[32][33];
    int b  = blockIdx.z;
    int n0 = blockIdx.x * 32;
    int d0 = blockIdx.y * 32;
    int tx = threadIdx.x, ty = threadIdx.y;          // block (32,8)
    const float* src = hr  + (size_t)b * DD * NN;
    float*       dst = hfT + (size_t)b * NN * DD;
    #pragma unroll
    for (int j = 0; j < 4; ++j)
        tile[ty + j * 8][tx] = src[(size_t)(d0 + ty + j * 8) * NN + n0 + tx];
    __syncthreads();
    #pragma unroll
    for (int j = 0; j < 4; ++j)
        dst[(size_t)(n0 + ty + j * 8) * DD + d0 + tx] = tile[tx][ty + j * 8];
}

// ---------------------------------------------------------------------------
// Kernel 5: fused rescorer: pair features -> WMMA MLP -> gelu -> resid ->
// softmax(log prior + resid) -> b128 weighted gather. 1 block per (b,n).
// ---------------------------------------------------------------------------
__device__ __forceinline__ float gelu_tanh(float x) {
    float x3 = x * x * x;
    float t  = tanhf(0.7978845608028654f * (x + 0.044715f * x3));
    return 0.5f * x * (1.f + t);
}

__global__ __launch_bounds__(128) void rescore_kernel(
    const float* __restrict__ qEmb, const float* __restrict__ kEmb,
    const float* __restrict__ prior, const int* __restrict__ idxg,
    const float* __restrict__ W1i, const float* __restrict__ W1,
    const float* __restrict__ b1,
    const float* __restrict__ W2, const float* __restrict__ b2,
    const float* __restrict__ hfT, float* __restrict__ outp)
{
    __shared__ float pair[16 * 516];   // 16 x 512, row pad->516 (stride%64==4)
    __shared__ float hbuf[16 * 68];
    __shared__ float qv[128];
    __shared__ float red[128];
    __shared__ float pri[16];
    __shared__ int   idk[16];
    __shared__ float wsm[16];

    int bn  = blockIdx.x;
    int b   = bn >> 12;
    int tid = threadIdx.x;

    if (tid < 16) {
        pri[tid] = prior[(size_t)bn * KK + tid];
        idk[tid] = idxg[(size_t)bn * KK + tid];
    }
    qv[tid] = qEmb[(size_t)bn * EE + tid];
    __syncthreads();

    float qq = qv[tid];
    for (int k = 0; k < KK; ++k) {
        float ks = kEmb[((size_t)b * NN + idk[k]) * EE + tid];
        float* pr = pair + k * 516;
        pr[tid]       = qq;
        pr[128 + tid] = ks;
        pr[256 + tid] = qq - ks;
        pr[384 + tid] = qq * ks;
    }
    __syncthreads();

    // h(16x64) = pair(16x512) @ W1[0:512] via f32 WMMA; prior column analytic.
    int lane = tid & 31, wave = tid >> 5;
    int half = lane >> 4, ml = lane & 15;
    int e0   = wave * 16;                 // 4 waves cover H=64
    const float* Wb = W1i + (size_t)(e0 + ml) * 2;
    v8f acc = {};
    for (int kk = 0; kk < 128; ++kk) {
        int k0 = kk * 4 + 2 * half;
        int kp = kk * 2 + half;
        v2f a = *(const v2f*)&pair[ml * 516 + k0];            // ds_load_b64
        v2f w = *(const v2f*)(Wb + (size_t)kp * HH * 2);      // global b64
        acc = __builtin_amdgcn_wmma_f32_16x16x4_f32(false, a, false, w,
                                                    (short)0, acc, false, false);
    }
    #pragma unroll
    for (int r = 0; r < 8; ++r) {
        int row = r + 8 * half;
        int e   = e0 + ml;
        float vv = acc[r] + pri[row] * W1[512 * HH + e] + b1[e];
        hbuf[row * 68 + e] = gelu_tanh(vv);
    }
    __syncthreads();

    // resid = h @ W2 + b2 ; logits = log(max(prior,1e-8)) + resid
    {
        int k = tid >> 3, i = tid & 7;
        float part = 0.f;
        #pragma unroll
        for (int j = 0; j < 8; ++j) part += hbuf[k * 68 + i * 8 + j] * W2[i * 8 + j];
        red[tid] = part;
    }
    __syncthreads();
    if (tid < 16) {
        float s = 0.f;
        #pragma unroll
        for (int j = 0; j < 8; ++j) s += red[tid * 8 + j];
        wsm[tid] = logf(fmaxf(pri[tid], 1e-8f)) + s + b2[0];
    }
    __syncthreads();
    if (tid == 0) {                       // softmax over 16 (trivial)
        float m = wsm[0];
        for (int j = 1; j < 16; ++j) m = fmaxf(m, wsm[j]);
        float se = 0.f, e_[16];
        for (int j = 0; j < 16; ++j) { e_[j] = __expf(wsm[j] - m); se += e_[j]; }
        float inv = 1.f / se;
        for (int j = 0; j < 16; ++j) wsm[j] = e_[j] * inv;
    }
    __syncthreads();

    // out[bn][:] = sum_k w[k] * hfT[b][idx_k][:] ; b128 loads/stores, L2-hot
    int d0 = tid * 8;
    v4f a0 = {0.f, 0.f, 0.f, 0.f}, a1 = {0.f, 0.f, 0.f, 0.f};
    for (int kq = 0; kq < KK; ++kq) {
        const float* rp = hfT + ((size_t)b * NN + idk[kq]) * DD + d0;
        if (kq + 1 < KK)
            __builtin_prefetch(hfT + ((size_t)b * NN + idk[kq + 1]) * DD + d0, 0, 1);
        float wv = wsm[kq];
        const v4f* r4 = (const v4f*)rp;
        a0 += wv * r4[0];
        a1 += wv * r4[1];
    }
    float* op = outp + (size_t)bn * DD + d0;
    *(v4f*)op       = a0;
    *((v4f*)op + 1) = a1;
}

// ---------------------------------------------------------------------------
extern "C" void kernel_launch(void* const* d_in, const int* in_sizes, int n_in,
                              void* d_out, int out_size, void* d_ws, size_t ws_size,
                              hipStream_t stream) {
    (void)in_sizes; (void)n_in; (void)out_size; (void)ws_size;
    const float* hr_attn = (const float*)d_in[0];
    const float* hr_hf   = (const float*)d_in[1];
    const float* base_hf = (const float*)d_in[2];
    const float* Wq = (const float*)d_in[3];
    const float* bq = (const float*)d_in[4];
    const float* Wk = (const float*)d_in[5];
    const float* bk = (const float*)d_in[6];
    const float* W1 = (const float*)d_in[7];
    const float* b1 = (const float*)d_in[8];
    const float* W2 = (const float*)d_in[9];
    const float* b2 = (const float*)d_in[10];
    float* outp = (float*)d_out;

    // workspace carve-up (floats)
    float* poolK = (float*)d_ws;                         // B*256*N
    float* poolQ = poolK + (size_t)BB * DIN * NN;
    float* kEmb  = poolQ + (size_t)BB * DIN * NN;        // B*N*E
    float* qEmb  = kEmb  + (size_t)BB * NN * EE;
    float* prior = qEmb  + (size_t)BB * NN * EE;         // B*N*K
    int*   idx   = (int*)(prior + (size_t)BB * NN * KK);
    float* hfT   = (float*)(idx + (size_t)BB * NN * KK); // B*N*D
    float* Wki   = hfT + (size_t)BB * NN * DD;           // 256*128
    float* Wqi   = Wki + (size_t)DIN * EE;
    float* W1i   = Wqi + (size_t)DIN * EE;               // 512*64

    interleave_w<<<dim3(64, 3), 256, 0, stream>>>(Wk, Wq, W1, Wki, Wqi, W1i);
    pool_kernel<<<dim3((BB * (DIN / 2) * NN) / 256, 2), 256, 0, stream>>>(hr_hf, base_hf, poolK, poolQ);
    encode_gemm<<<dim3((BB * NN) / 16, 2), 256, 0, stream>>>(poolK, poolQ, Wki, bk, Wqi, bq, kEmb, qEmb);
    topk_kernel<<<BB * NN, 256, 0, stream>>>(hr_attn, prior, idx);
    transpose_hf<<<dim3(NN / 32, DD / 32, BB), dim3(32, 8), 0, stream>>>(hr_hf, hfT);
    rescore_kernel<<<BB * NN, 128, 0, stream>>>(qEmb, kEmb, prior, idx, W1i, W1, b1, W2, b2, hfT, outp);
}